// cross_att_75453985456319
// MI455X (gfx1250) — compile-verified
//
#include <hip/hip_runtime.h>

// ---------------------------------------------------------------------------
// cross_att for B=4, S=1024, D_MODEL=32, H=32 (per-head dim = 1).
// Math: softmax_j(Q_i + K_j) == softmax_j(K_j)  => output independent of Q/x2,
// and constant across the sequence dimension. See analysis above.
// ---------------------------------------------------------------------------

typedef float v2f __attribute__((ext_vector_type(2)));
typedef float v8f __attribute__((ext_vector_type(8)));

#define S_LEN 1024
#define NH    32
#define DM    32
#define NSC   8      // s-chunks per (b, head-tile)
#define NWAVE 8      // waves per block (wave32)

__device__ __forceinline__ void online_merge(float& m, float& z, float& w,
                                             float m2, float z2, float w2) {
    float M = fmaxf(m, m2);
    float a = expf(m - M);
    float b = expf(m2 - M);
    m = M;
    z = z * a + z2 * b;
    w = w * a + w2 * b;
}

// Kernel 1: per wave, compute a 16(s) x 16(h) tile of K and V with fp32 WMMA,
// then online-softmax-reduce over s within the block; emit one partial
// (max, sumexp, sumexp*V) triple per (b, h, s-chunk).
__global__ __launch_bounds__(256) void kv_softmax_partial(
    const float* __restrict__ x,
    const float* __restrict__ Wk, const float* __restrict__ bk,
    const float* __restrict__ Wv, const float* __restrict__ bv,
    float* __restrict__ pm, float* __restrict__ pz, float* __restrict__ pw)
{
    const int sc  = blockIdx.x;   // 0..7   s-chunk (128 rows)
    const int ht  = blockIdx.y;   // 0..1   head tile of 16
    const int b   = blockIdx.z;   // 0..3
    const int tid = threadIdx.x;
    const int wave = tid >> 5;
    const int lane = tid & 31;
    const int l15  = lane & 15;
    const int hi2  = (lane >> 4) * 2;          // 0 or 2 (K offset for A/B frags)
    const int s0   = (sc * NWAVE + wave) * 16; // 16-row tile start

    // V_WMMA_F32_16X16X4_F32 fragment layouts (wave32):
    //   A (16x4):  lane l holds M = l&15, K = {r + (l<16?0:2)}, r=0..1
    //   B (4x16):  lane l holds N = l&15, K = {r + (l<16?0:2)}, r=0..1
    //   C/D:       lane l, reg r -> M = r + (l<16?0:8), N = l&15
    const float* xrow  = x  + ((size_t)b * S_LEN + s0 + l15) * DM + hi2;
    const float* wkrow = Wk + (size_t)(ht * 16 + l15) * DM + hi2;  // B = Wk^T
    const float* wvrow = Wv + (size_t)(ht * 16 + l15) * DM + hi2;  // B = Wv^T

    v8f accK = {};
    v8f accV = {};
#pragma unroll
    for (int kc = 0; kc < 8; ++kc) {           // K = 32 in chunks of 4
        v2f a  = *(const v2f*)(xrow  + kc * 4);
        v2f bK = *(const v2f*)(wkrow + kc * 4);
        v2f bV = *(const v2f*)(wvrow + kc * 4);
        accK = __builtin_amdgcn_wmma_f32_16x16x4_f32(
                   false, a, false, bK, (short)0, accK, false, false);
        accV = __builtin_amdgcn_wmma_f32_16x16x4_f32(
                   false, a, false, bV, (short)0, accV, false, false);
    }

    const int h = ht * 16 + l15;               // this lane's head column
    const float biasK = bk[h];
    const float biasV = bv[h];

    // Online softmax over the 8 s-values this lane holds (column h fixed).
    float m = -3.0e38f, z = 0.0f, w = 0.0f;
#pragma unroll
    for (int r = 0; r < 8; ++r) {
        float kv = accK[r] + biasK;
        float vv = accV[r] + biasV;
        float M  = fmaxf(m, kv);
        float sA = expf(m - M);
        float e  = expf(kv - M);
        z = z * sA + e;
        w = w * sA + e * vv;
        m = M;
    }

    // Block-level merge: lanes {h, h+16} of all 8 waves share head h.
    __shared__ float sm_[256], sz_[256], sw_[256];
    sm_[tid] = m; sz_[tid] = z; sw_[tid] = w;
    __syncthreads();

    if (tid < 16) {
        float M = sm_[tid], Z = sz_[tid], W = sw_[tid];
        online_merge(M, Z, W, sm_[tid + 16], sz_[tid + 16], sw_[tid + 16]);
#pragma unroll
        for (int wv = 1; wv < NWAVE; ++wv) {
            online_merge(M, Z, W, sm_[wv * 32 + tid],      sz_[wv * 32 + tid],      sw_[wv * 32 + tid]);
            online_merge(M, Z, W, sm_[wv * 32 + tid + 16], sz_[wv * 32 + tid + 16], sw_[wv * 32 + tid + 16]);
        }
        const int idx = (b * NH + (ht * 16 + tid)) * NSC + sc;
        pm[idx] = M; pz[idx] = Z; pw[idx] = W;
    }
}

// Kernel 2: merge chunk partials -> c[b,h]; project one row per batch:
// yrow[b,d] = sum_h c[b,h] * Wo[d,h] + bo[d]
__global__ __launch_bounds__(128) void merge_project(
    const float* __restrict__ pm, const float* __restrict__ pz,
    const float* __restrict__ pw,
    const float* __restrict__ Wo, const float* __restrict__ bo,
    float* __restrict__ yrow)
{
    const int t = threadIdx.x;                 // 0..127 == (b*32 + h)
    float M = pm[t * NSC], Z = pz[t * NSC], W = pw[t * NSC];
#pragma unroll
    for (int sc = 1; sc < NSC; ++sc)
        online_merge(M, Z, W, pm[t * NSC + sc], pz[t * NSC + sc], pw[t * NSC + sc]);

    __shared__ float c[128];
    c[t] = W / Z;
    __syncthreads();

    const int b = t >> 5, d = t & 31;
    float acc = bo[d];
#pragma unroll
    for (int h = 0; h < NH; ++h)
        acc += c[b * NH + h] * Wo[d * DM + h];
    yrow[t] = acc;                             // yrow[b*32 + d]
}

// Kernel 3: out[b,s,:] = yrow[b,:] for all s (the only real HBM traffic).
__global__ __launch_bounds__(256) void broadcast_rows(
    const float* __restrict__ yrow, float* __restrict__ out)
{
    const int gid = blockIdx.x * blockDim.x + threadIdx.x; // one float4 each
    const int e   = gid * 4;                   // flat element index
    const int b   = e >> 15;                   // S*DM = 32768 elements/batch
    const int d   = e & (DM - 1);
    const float* yr = yrow + b * DM + d;
    float4 v = make_float4(yr[0], yr[1], yr[2], yr[3]);
    *(float4*)(out + e) = v;
}

extern "C" void kernel_launch(void* const* d_in, const int* in_sizes, int n_in,
                              void* d_out, int out_size, void* d_ws, size_t ws_size,
                              hipStream_t stream) {
    (void)in_sizes; (void)n_in; (void)out_size; (void)ws_size;
    // setup_inputs order: x, x2, Wq, bq, Wk, bk, Wv, bv, Wo, bo
    const float* x  = (const float*)d_in[0];
    const float* Wk = (const float*)d_in[4];
    const float* bk = (const float*)d_in[5];
    const float* Wv = (const float*)d_in[6];
    const float* bv = (const float*)d_in[7];
    const float* Wo = (const float*)d_in[8];
    const float* bo = (const float*)d_in[9];
    float* out = (float*)d_out;

    // Workspace layout: 3 partial arrays [4*32*8] + yrow[128]  (~13 KB)
    float* pm   = (float*)d_ws;
    float* pz   = pm + 4 * NH * NSC;
    float* pw   = pz + 4 * NH * NSC;
    float* yrow = pw + 4 * NH * NSC;

    dim3 grid1(NSC, 2, 4);
    kv_softmax_partial<<<grid1, 256, 0, stream>>>(x, Wk, bk, Wv, bv, pm, pz, pw);
    merge_project<<<1, 128, 0, stream>>>(pm, pz, pw, Wo, bo, yrow);
    broadcast_rows<<<4 * S_LEN * DM / 4 / 256, 256, 0, stream>>>(yrow, out);
}